// conv2d_69458211111710
// MI455X (gfx1250) — compile-verified
//
#include <hip/hip_runtime.h>
#include <hip/hip_bf16.h>

// ---------------------------------------------------------------------------
// leaky( conv3x3(x - parity_mean) + bias ) + parity_mean  on MI455X (gfx1250)
//
//   out[b,o,i,j] = sum_{c,r,s} w[o,c,r,s] * xpad[b,c, i+s, j+r]   (edge pad)
//   conv(x - avg_par) = conv(x) - Wsum[o,0]*avg0 - Wsum[o,1]*avg1
//
// Implicit GEMM on V_WMMA_F32_16X16X4_F32 (wave32):
//   M = 16 output channels / wave, N = 16 pixels / tile (4 tiles / wave),
//   K = 128 channels x 9 taps = 1152 (4 channels per WMMA step).
// A frag (ISA 16x4 f32): lane L -> M = L&15, K = 2*(L>=16) + {0,1}
//   -> one aligned global b64 load from tap-major transposed weights.
// B frag: lane L -> N = L&15, K = 2*(L>=16) + {0,1}. LDS is stored
//   channel-PAIR interleaved ( [row][c/2][2*col + (c&1)] ) so the two K
//   values are adjacent -> ONE ds_load_b64 per fragment (1 DS issue : 1 WMMA).
//   Pair-row stride 160 dwords => half-wave groups 32 banks apart, stride-2
//   cols span the other 32 banks: conflict-free.
// Staging uses GLOBAL_LOAD_ASYNC_TO_LDS_B32 (ASYNCcnt): each wave issue
//   reads 128 contiguous global bytes and scatters stride-2 into LDS.
// ---------------------------------------------------------------------------

typedef __attribute__((ext_vector_type(2))) float v2f;
typedef __attribute__((ext_vector_type(8))) float v8f;

#define HW    64
#define CIN   128
#define COUT  128
#define SLOPE 0.01f

// ---- prep: Wt[t][o][c] = w[o][c][t%3][t/3]  (c contiguous for A-frag loads)
__global__ __launch_bounds__(128) void prep_wt(const float* __restrict__ w,
                                               float* __restrict__ wt) {
  const int o = blockIdx.x;
  const int c = threadIdx.x;
#pragma unroll
  for (int t = 0; t < 9; ++t) {
    const int r = t % 3, s = t / 3;
    wt[((size_t)t * COUT + o) * CIN + c] = w[((o * CIN + c) * 3 + r) * 3 + s];
  }
}

// ---- prep: Wsum[o][u] = sum_{c%2==u} sum_{r,s} w[o,c,r,s]
__global__ __launch_bounds__(128) void prep_wsum(const float* __restrict__ w,
                                                 float* __restrict__ wsum) {
  const int o = threadIdx.x;
  float s0 = 0.f, s1 = 0.f;
  for (int c = 0; c < CIN; ++c) {
    float a = 0.f;
#pragma unroll
    for (int t = 0; t < 9; ++t) a += w[(o * CIN + c) * 9 + t];
    if (c & 1) s1 += a; else s0 += a;
  }
  wsum[o * 2 + 0] = s0;
  wsum[o * 2 + 1] = s1;
}

// async 4B/lane copy: from global address sbase + goff to the LDS address in lds_addr   (GVS mode, ASYNCcnt)
__device__ __forceinline__ void async_ld_b32(unsigned lds_addr, unsigned goff,
                                             unsigned long long sbase) {
  asm volatile("global_load_async_to_lds_b32 %0, %1, %2"
               :
               : "v"(lds_addr), "v"(goff), "s"(sbase)
               : "memory");
}

__device__ __forceinline__ void wait_async0() {
  asm volatile("s_wait_asynccnt 0x0" ::: "memory");
}

// ---- main: one workgroup per (batch, output row); 8 waves cover 128 o-chans
__global__ __launch_bounds__(256) void conv_wmma(const float* __restrict__ x,
                                                 const float* __restrict__ wt,
                                                 const float* __restrict__ wsum,
                                                 const float* __restrict__ bias,
                                                 float* __restrict__ out) {
  // (row, chan c, col) lives at ldsx[row][c>>1][2*col + (c&1)]
  __shared__ float ldsx[3][32][160];      // 61,440 B
  __shared__ float colsum_s[2][64];       // parity column sums over 3 rows
  __shared__ float avg_s[2][64];          // parity means (after 3-col box)

  const int i    = blockIdx.x;            // output row
  const int b    = blockIdx.y;            // batch
  const int tid  = threadIdx.x;
  const int lane = tid & 31;
  const int wave = tid >> 5;
  const int obase = wave * 16;            // this wave's 16 output channels
  const int mn   = lane & 15;             // M (A) == N (B/C) index for lane
  const int hi   = lane >> 4;             // 0: lanes 0-15, 1: lanes 16-31
  const int koff = hi << 1;               // K offset 0 or 2 inside 4-step

  // clamped source rows (edge padding)
  const int y0 = (i > 0) ? i - 1 : 0;
  const int y2 = (i < HW - 1) ? i + 1 : HW - 1;
  const int yrow[3] = {y0, i, y2};

  // clamped columns per (tile, tap-col r)
  int cols[4][3];
#pragma unroll
  for (int tile = 0; tile < 4; ++tile)
#pragma unroll
    for (int r = 0; r < 3; ++r) {
      int cc = tile * 16 + mn + r - 1;
      cols[tile][r] = cc < 0 ? 0 : (cc > HW - 1 ? HW - 1 : cc);
    }

  v8f acc[4] = {v8f(0.f), v8f(0.f), v8f(0.f), v8f(0.f)};
  float cs = 0.f;                         // partial parity column sum
  const int ux = tid & 63;
  const int uu = (tid >> 6) & 1;

  const unsigned ldsbase = (unsigned)(uintptr_t)&ldsx[0][0][0];
  const unsigned long long xbase = (unsigned long long)(uintptr_t)x;

  for (int phase = 0; phase < 2; ++phase) {
    // ---- async-stage 3 rows x 64 chans x 64 cols (4B/lane, ASYNCcnt) ----
    // idx -> col = idx&63 (contiguous across a wave), c = (idx>>6)&63,
    // row = idx>>12; each wave issue reads 128 contiguous global bytes.
#pragma unroll 4
    for (int k = 0; k < 48; ++k) {
      const int idx = tid + k * 256;      // 0..12287
      const int col = idx & 63;
      const int c   = (idx >> 6) & 63;
      const int row = idx >> 12;
      const unsigned goff = (unsigned)(
          (((b * CIN + phase * 64 + c) * HW + yrow[row]) * HW + col) * 4);
      const unsigned laddr = ldsbase +
          (unsigned)((((row * 32 + (c >> 1)) * 160) + 2 * col + (c & 1)) * 4);
      async_ld_b32(laddr, goff, xbase);
    }
    wait_async0();
    __syncthreads();

    // ---- parity column sums over this phase (global chan parity == c&1)
    if (tid < 128) {
      float s = 0.f;
#pragma unroll
      for (int row = 0; row < 3; ++row)
        for (int cp = 0; cp < 32; ++cp) s += ldsx[row][cp][2 * ux + uu];
      cs += s;
    }

    // ---- WMMA K-loop: 9 taps x 16 channel-steps of 4
#pragma unroll
    for (int t = 0; t < 9; ++t) {
      const int sr = t / 3;               // row offset (LDS row index)
      const int rc = t % 3;               // col offset
      const float* wrow = wt +
          (((size_t)t * COUT + obase + mn) * CIN + phase * 64 + koff);
      for (int cl = 0; cl < 64; cl += 4) {
        const v2f a = *(const v2f*)(wrow + cl);   // A frag: K=koff, koff+1
        const int cp = (cl >> 1) + hi;            // pair row: chans cl+koff..+1
#pragma unroll
        for (int tile = 0; tile < 4; ++tile) {
          const int col = cols[tile][rc];
          const v2f bb = *(const v2f*)&ldsx[sr][cp][2 * col];  // ds_load_b64
          acc[tile] = __builtin_amdgcn_wmma_f32_16x16x4_f32(
              false, a, false, bb, (short)0, acc[tile], false, false);
        }
      }
    }
    __syncthreads();                      // all LDS reads done before restage
  }

  // ---- parity means: 3-col clamped box of column sums, / 576
  if (tid < 128) colsum_s[uu][ux] = cs;
  __syncthreads();
  if (tid < 128) {
    const int xm = ux > 0 ? ux - 1 : 0;
    const int xp = ux < HW - 1 ? ux + 1 : HW - 1;
    avg_s[uu][ux] =
        (colsum_s[uu][xm] + colsum_s[uu][ux] + colsum_s[uu][xp]) * (1.f / 576.f);
  }
  __syncthreads();

  // ---- epilogue: bias - Wsum0*avg0 - Wsum1*avg1, leaky, + avg[o%2]
  float bv[8], w0[8], w1[8];
#pragma unroll
  for (int v = 0; v < 8; ++v) {
    const int o = obase + v + (hi << 3);  // C/D layout: VGPR v -> rows v, v+8
    bv[v] = bias[o];
    w0[v] = wsum[o * 2 + 0];
    w1[v] = wsum[o * 2 + 1];
  }
#pragma unroll
  for (int tile = 0; tile < 4; ++tile) {
    const int j = tile * 16 + mn;
    const float a0 = avg_s[0][j], a1 = avg_s[1][j];
#pragma unroll
    for (int v = 0; v < 8; ++v) {
      const int o = obase + v + (hi << 3);
      float val = acc[tile][v] + bv[v] - w0[v] * a0 - w1[v] * a1;
      val = val > 0.f ? val : SLOPE * val;
      val += (v & 1) ? a1 : a0;           // o parity == v parity here
      out[(((size_t)b * COUT + o) * HW + i) * HW + j] = val;
    }
  }
}

extern "C" void kernel_launch(void* const* d_in, const int* in_sizes, int n_in,
                              void* d_out, int out_size, void* d_ws,
                              size_t ws_size, hipStream_t stream) {
  const float* xx   = (const float*)d_in[0];   // (16,128,64,64) f32
  const float* w    = (const float*)d_in[1];   // (128,128,3,3)  f32
  const float* bias = (const float*)d_in[2];   // (128,)         f32
  float* out = (float*)d_out;                  // (16,128,64,64) f32

  float* wt   = (float*)d_ws;                  // 9*128*128 floats
  float* wsum = wt + 9 * COUT * CIN;           // 256 floats

  prep_wt<<<COUT, CIN, 0, stream>>>(w, wt);
  prep_wsum<<<1, COUT, 0, stream>>>(w, wsum);
  conv_wmma<<<dim3(HW, 16), 256, 0, stream>>>(xx, wt, wsum, bias, out);
}